// TokenAttentionLayer_24979529794141
// MI455X (gfx1250) — compile-verified
//
#include <hip/hip_runtime.h>
#include <hip/hip_bf16.h>

typedef __attribute__((ext_vector_type(2))) float v2f;
typedef __attribute__((ext_vector_type(8))) float v8f;

#define ALPHA 0.2f
#define NEG_BIG -9000000000000000.0f

// Block tile 128x128, K-tile 32, 8 waves (4 along M x 2 along N),
// each wave: 32(M) x 64(N) = 8 v8f accumulators.
#define TM 128
#define TN 128
#define TK 32
#define LDT 36  /* tile row stride in floats: mult of 4 (b128 stores), 36*l%64 distinct -> conflict-free b64 frag reads */

__device__ __forceinline__ float leaky(float x) { return x >= 0.0f ? x : ALPHA * x; }

// ---------------------------------------------------------------------------
// Kernel 1: Wh = h @ W  (M = B*N, K = Fin, Nc = Fout), f32 WMMA, double-buffered
// Tiles stored [row][k]: A as [i][k] (natural), B transposed to [f][k].
// ---------------------------------------------------------------------------
__global__ __launch_bounds__(256) void gemm_h_W(const float* __restrict__ A,
                                                const float* __restrict__ Bm,
                                                float* __restrict__ C,
                                                int Nc, int K) {
  __shared__ float sA[2][TM * LDT];
  __shared__ float sB[2][TN * LDT];

  const int tid  = threadIdx.x;
  const int lane = tid & 31;
  const int w    = tid >> 5;
  const int wi   = w & 3;    // 4 waves along M (32 rows each)
  const int wf   = w >> 2;   // 2 waves along N (64 cols each)
  const int half = lane >> 4;
  const int l    = lane & 15;
  const int iBase = blockIdx.y * TM;
  const int fBase = blockIdx.x * TN;

  v8f acc[2][4];
  for (int q = 0; q < 2; ++q)
    for (int t = 0; t < 4; ++t)
      for (int r = 0; r < 8; ++r) acc[q][t][r] = 0.0f;

  float4 aregs[4];   // prefetch staging: A tile 128x32 -> 4 float4/thread
  float  bregs[16];  // prefetch staging: B tile 32x128 -> 16 floats/thread (transposed)

  auto loadA = [&](int k0) {
    #pragma unroll
    for (int s = 0; s < 4; ++s) {
      int e = s * 256 + tid;
      int kq = e & 7, ii = e >> 3;
      aregs[s] = *(const float4*)&A[(size_t)(iBase + ii) * K + k0 + kq * 4];
    }
  };
  auto loadB = [&](int k0) {
    #pragma unroll
    for (int s = 0; s < 16; ++s) {
      int e = s * 256 + tid;
      int f = e & 127, kk = e >> 7;
      bregs[s] = Bm[(size_t)(k0 + kk) * Nc + fBase + f];
    }
  };
  auto storeT = [&](int buf) {
    #pragma unroll
    for (int s = 0; s < 4; ++s) {
      int e = s * 256 + tid;
      int kq = e & 7, ii = e >> 3;
      *(float4*)&sA[buf][ii * LDT + kq * 4] = aregs[s];
    }
    #pragma unroll
    for (int s = 0; s < 16; ++s) {
      int e = s * 256 + tid;
      int f = e & 127, kk = e >> 7;
      sB[buf][f * LDT + kk] = bregs[s];   // transpose: [f][k]
    }
  };
  auto compute = [&](int buf) {
    #pragma unroll
    for (int ks = 0; ks < TK; ks += 4) {
      v2f af[2];
      #pragma unroll
      for (int q = 0; q < 2; ++q)
        af[q] = *(const v2f*)&sA[buf][(wi * 32 + q * 16 + l) * LDT + ks + 2 * half];
      #pragma unroll
      for (int t = 0; t < 4; ++t) {
        v2f bf = *(const v2f*)&sB[buf][(wf * 64 + t * 16 + l) * LDT + ks + 2 * half];
        #pragma unroll
        for (int q = 0; q < 2; ++q)
          acc[q][t] = __builtin_amdgcn_wmma_f32_16x16x4_f32(
              false, af[q], false, bf, (short)0, acc[q][t], false, false);
      }
    }
  };

  const int T = K / TK;
  loadA(0); loadB(0); storeT(0);
  __syncthreads();
  for (int t = 0; t < T; ++t) {
    if (t + 1 < T) { loadA((t + 1) * TK); loadB((t + 1) * TK); }
    compute(t & 1);
    __syncthreads();
    if (t + 1 < T) { storeT((t + 1) & 1); __syncthreads(); }
  }

  for (int q = 0; q < 2; ++q) {
    const int ib = iBase + wi * 32 + q * 16 + 8 * half;
    for (int t = 0; t < 4; ++t) {
      const int fo = fBase + wf * 64 + t * 16 + l;
      for (int r = 0; r < 8; ++r)
        C[(size_t)(ib + r) * Nc + fo] = acc[q][t][r];
    }
  }
}

// ---------------------------------------------------------------------------
// Kernel 2: s_i[row] = Wh[row]·a1 ; s_j[row] = Wh[row]·a2   (wave per row)
// ---------------------------------------------------------------------------
__global__ __launch_bounds__(256) void row_scores(const float* __restrict__ Wh,
                                                  const float* __restrict__ a,
                                                  float* __restrict__ si,
                                                  float* __restrict__ sj,
                                                  int Fo) {
  const int row  = blockIdx.x * 8 + (threadIdx.x >> 5);
  const int lane = threadIdx.x & 31;
  const float* wr = Wh + (size_t)row * Fo;
  const float* a1 = a;
  const float* a2 = a + Fo;
  float s1 = 0.0f, s2 = 0.0f;
  for (int f = lane; f < Fo; f += 32) {
    float wv = wr[f];
    s1 += wv * a1[f];
    s2 += wv * a2[f];
  }
  for (int m = 16; m >= 1; m >>= 1) {
    s1 += __shfl_xor(s1, m, 32);
    s2 += __shfl_xor(s2, m, 32);
  }
  if (lane == 0) { si[row] = s1; sj[row] = s2; }
}

// ---------------------------------------------------------------------------
// Kernel 3: per-column online softmax stats  m_j, 1/Z_j  (softmax over i)
// ---------------------------------------------------------------------------
__global__ __launch_bounds__(256) void col_stats(const int* __restrict__ adj,
                                                 const float* __restrict__ si,
                                                 const float* __restrict__ sj,
                                                 float* __restrict__ mcol,
                                                 float* __restrict__ zrcp,
                                                 int Nn) {
  const int b = blockIdx.y;
  const int j = blockIdx.x * blockDim.x + threadIdx.x;
  const int* adjB = adj + (size_t)b * Nn * Nn;
  const float* siB = si + (size_t)b * Nn;
  const float sjv = sj[(size_t)b * Nn + j];

  float m = -3.0e38f;
  float z = 0.0f;
  for (int i = 0; i < Nn; ++i) {
    int   ad = adjB[(size_t)i * Nn + j];      // coalesced across j
    float v  = (ad > 0) ? leaky(siB[i] + sjv) : NEG_BIG;
    if (v > m) { z = z * __expf(m - v) + 1.0f; m = v; }
    else       { z += __expf(v - m); }
  }
  mcol[(size_t)b * Nn + j] = m;
  zrcp[(size_t)b * Nn + j] = 1.0f / z;
}

// ---------------------------------------------------------------------------
// Kernel 4: out = leaky( P @ Wh ), P generated on the fly (overlapped with WMMA)
// ---------------------------------------------------------------------------
__global__ __launch_bounds__(256) void attn_gemm(const int* __restrict__ adj,
                                                 const float* __restrict__ Wh,
                                                 const float* __restrict__ si,
                                                 const float* __restrict__ sj,
                                                 const float* __restrict__ mcol,
                                                 const float* __restrict__ zrcp,
                                                 float* __restrict__ out,
                                                 int Nn, int Fo) {
  __shared__ float sA[2][TM * LDT];
  __shared__ float sB[2][TN * LDT];

  const int b    = blockIdx.z;
  const int tid  = threadIdx.x;
  const int lane = tid & 31;
  const int w    = tid >> 5;
  const int wi   = w & 3;
  const int wf   = w >> 2;
  const int half = lane >> 4;
  const int l    = lane & 15;
  const int iBase = blockIdx.y * TM;
  const int fBase = blockIdx.x * TN;

  const int*   adjB = adj + (size_t)b * Nn * Nn;
  const float* WhB  = Wh + (size_t)b * Nn * Fo;
  const float* siB  = si + (size_t)b * Nn;
  const float* sjB  = sj + (size_t)b * Nn;
  const float* mB   = mcol + (size_t)b * Nn;
  const float* zB   = zrcp + (size_t)b * Nn;
  float*       outB = out + (size_t)b * Nn * Fo;

  v8f acc[2][4];
  for (int q = 0; q < 2; ++q)
    for (int t = 0; t < 4; ++t)
      for (int r = 0; r < 8; ++r) acc[q][t][r] = 0.0f;

  float pregs[16];   // generated attention tile: 128x32 -> 16/thread
  float bregs[16];   // Wh tile (transposed): 32x128 -> 16/thread

  auto genA = [&](int k0) {
    #pragma unroll
    for (int s = 0; s < 16; ++s) {
      int e  = s * 256 + tid;
      int kk = e & 31, ii = e >> 5;
      int gi = iBase + ii, gj = k0 + kk;
      int   ad = adjB[(size_t)gi * Nn + gj];     // coalesced in j
      float v  = (ad > 0) ? leaky(siB[gi] + sjB[gj]) : NEG_BIG;
      pregs[s] = __expf(v - mB[gj]) * zB[gj];
    }
  };
  auto loadB = [&](int k0) {
    #pragma unroll
    for (int s = 0; s < 16; ++s) {
      int e = s * 256 + tid;
      int f = e & 127, kk = e >> 7;
      bregs[s] = WhB[(size_t)(k0 + kk) * Fo + fBase + f];
    }
  };
  auto storeT = [&](int buf) {
    #pragma unroll
    for (int s = 0; s < 16; ++s) {
      int e  = s * 256 + tid;
      int kk = e & 31, ii = e >> 5;
      sA[buf][ii * LDT + kk] = pregs[s];
    }
    #pragma unroll
    for (int s = 0; s < 16; ++s) {
      int e = s * 256 + tid;
      int f = e & 127, kk = e >> 7;
      sB[buf][f * LDT + kk] = bregs[s];   // transpose: [f][k]
    }
  };
  auto compute = [&](int buf) {
    #pragma unroll
    for (int ks = 0; ks < TK; ks += 4) {
      v2f af[2];
      #pragma unroll
      for (int q = 0; q < 2; ++q)
        af[q] = *(const v2f*)&sA[buf][(wi * 32 + q * 16 + l) * LDT + ks + 2 * half];
      #pragma unroll
      for (int t = 0; t < 4; ++t) {
        v2f bf = *(const v2f*)&sB[buf][(wf * 64 + t * 16 + l) * LDT + ks + 2 * half];
        #pragma unroll
        for (int q = 0; q < 2; ++q)
          acc[q][t] = __builtin_amdgcn_wmma_f32_16x16x4_f32(
              false, af[q], false, bf, (short)0, acc[q][t], false, false);
      }
    }
  };

  const int T = Nn / TK;
  genA(0); loadB(0); storeT(0);
  __syncthreads();
  for (int t = 0; t < T; ++t) {
    if (t + 1 < T) { genA((t + 1) * TK); loadB((t + 1) * TK); }
    compute(t & 1);
    __syncthreads();
    if (t + 1 < T) { storeT((t + 1) & 1); __syncthreads(); }
  }

  for (int q = 0; q < 2; ++q) {
    const int ib = iBase + wi * 32 + q * 16 + 8 * half;
    for (int t = 0; t < 4; ++t) {
      const int fo = fBase + wf * 64 + t * 16 + l;
      for (int r = 0; r < 8; ++r)
        outB[(size_t)(ib + r) * Fo + fo] = leaky(acc[q][t][r]);
    }
  }
}

// ---------------------------------------------------------------------------
extern "C" void kernel_launch(void* const* d_in, const int* in_sizes, int n_in,
                              void* d_out, int out_size, void* d_ws, size_t ws_size,
                              hipStream_t stream) {
  const int B = 4, N = 2048, Fin = 512, Fout = 512;

  const float* h   = (const float*)d_in[0];
  const float* W   = (const float*)d_in[1];
  const float* a   = (const float*)d_in[2];
  const int*   adj = (const int*)d_in[3];
  float* out = (float*)d_out;

  // workspace layout
  float* Wh = (float*)d_ws;                       // B*N*Fout
  float* si = Wh + (size_t)B * N * Fout;          // B*N
  float* sj = si + (size_t)B * N;                 // B*N
  float* mc = sj + (size_t)B * N;                 // B*N
  float* zr = mc + (size_t)B * N;                 // B*N

  // 1) Wh = h @ W
  {
    dim3 grid(Fout / TN, (B * N) / TM);
    gemm_h_W<<<grid, 256, 0, stream>>>(h, W, Wh, Fout, Fin);
  }
  // 2) s_i, s_j
  {
    dim3 grid((B * N) / 8);
    row_scores<<<grid, 256, 0, stream>>>(Wh, a, si, sj, Fout);
  }
  // 3) column softmax stats (softmax over axis=1 -> per-column over i)
  {
    dim3 grid(N / 256, B);
    col_stats<<<grid, 256, 0, stream>>>(adj, si, sj, mc, zr, N);
  }
  // 4) out = leaky(attention @ Wh), attention generated on the fly
  {
    dim3 grid(Fout / TN, N / TM, B);
    attn_gemm<<<grid, 256, 0, stream>>>(adj, Wh, si, sj, mc, zr, out, N, Fout);
  }
}